// GATv2Layer_7069516169233
// MI455X (gfx1250) — compile-verified
//
#include <hip/hip_runtime.h>
#include <hip/hip_bf16.h>
#include <math.h>

// ---------------------------------------------------------------------------
// GATv2 layer with built-in KNN graph, MI455X (gfx1250, wave32, WMMA + TDM).
//   N=16384 nodes, D=64 in-features, K=16 neighbors, HEADS=5, OUT=128.
// Pipeline:
//   1) sqnorm:   sq[i] = ||x_i||^2
//   2) feat:     xfeat[N][1280] = [x@W_l + b_l | x@W_r + b_r]   (fp32 WMMA)
//   3) knn:      fused x.x^T distance GEMM (fp32 WMMA 16x16x4, K=64)
//                + register-resident per-row top-16 selection.
//                Candidate tiles DMA'd into LDS by the Tensor Data Mover with
//                DOUBLE BUFFERING: tile ct+1 flies while tile ct is consumed
//                (s_wait_tensorcnt <= 1 exploits in-order TDM completion).
//                TDM pad (4 DW per 64 DW) gives the bank-conflict-free
//                stride-68 row layout for WMMA operand reads.
//   4) gat:      gather + leaky_relu + att dot + softmax(K) + weighted sum,
//                head mean + bias. One wave32 per node, __shfl_xor reduces.
// All heavy operands (x: 4MB, W: 0.33MB, xfeat: 84MB) are L2-resident (192MB),
// so HBM traffic is ~compulsory only; compute rides the WMMA pipe.
// ---------------------------------------------------------------------------

#define N_NODES   16384
#define D_IN      64
#define KNN_K     16
#define HEADS     5
#define OUT_C     128
#define F_TOT     (HEADS * OUT_C)        // 640
#define F2_TOT    (2 * F_TOT)            // 1280 (x_l | x_r concatenated)
#define NEG_SLOPE 0.2f

typedef float        v2f __attribute__((ext_vector_type(2)));
typedef float        v8f __attribute__((ext_vector_type(8)));
typedef unsigned int v4u __attribute__((ext_vector_type(4)));
typedef unsigned int v8u __attribute__((ext_vector_type(8)));

// fp32 WMMA: D(16x16,f32) = A(16x4,f32) * B(4x16,f32) + C
// 8 args: (neg_a, A, neg_b, B, c_mod, C, reuse_a, reuse_b)
__device__ __forceinline__ v8f wmma_f32_k4(v2f a, v2f b, v8f c) {
  return __builtin_amdgcn_wmma_f32_16x16x4_f32(false, a, false, b,
                                               (short)0, c, false, false);
}

// ---------------------------------------------------------------------------
// TDM: DMA a 64-row x 64-float tile (row-major, contiguous) from global into
// LDS, inserting 4 pad DWORDs after every 64 DWORDs -> LDS row stride of 68
// floats (bank-conflict-free for the WMMA b64 operand reads).
// D# encoding per CDNA5 ISA ch.7/8 (async_tensor):
//  group0: [1:0]count=1 | [63:32]lds_addr | [120:64]global_addr | [127:126]=2
//  group1: data_size=4B, pad_enable, pad_interval=5 (64 DW), pad_amount=3 (4 DW),
//          tensor_dim0=64, tensor_dim1=N, tile_dim0=64, tile_dim1=64,
//          tensor_dim0_stride=64.
// 2-D tensor -> descriptor groups 2/3 omitted (VADDR2/3 = NULL).
// ---------------------------------------------------------------------------
__device__ __forceinline__ void tdm_load_tile_64x64(const float* gsrc,
                                                    unsigned int lds_byte_off) {
  const unsigned long long ga = (unsigned long long)(uintptr_t)gsrc;
  v4u g0;
  g0[0] = 1u;                                        // count=1 (valid), user mode
  g0[1] = lds_byte_off;                              // lds_addr
  g0[2] = (unsigned int)(ga & 0xFFFFFFFFu);          // global_addr[31:0]
  g0[3] = (unsigned int)((ga >> 32) & 0x01FFFFFFu)   // global_addr[56:32]
          | (2u << 30);                              // type = 2 ("image")
  v8u g1;
  g1[0] = (2u << 16)      // data_size = 4 bytes
        | (1u << 20)      // pad_enable
        | (5u << 22)      // pad_interval: 64 DWORDs
        | (3u << 25);     // pad_amount:   4 DWORDs
  g1[1] = (64u & 0xFFFFu) << 16;               // tensor_dim0 = 64 (low 16)
  g1[2] = ((unsigned)N_NODES & 0xFFFFu) << 16; // dim0 hi=0 | tensor_dim1 lo16
  g1[3] = (64u << 16);                         // tensor_dim1 hi=0 | tile_dim0=64
  g1[4] = 64u;                                 // tile_dim1=64, tile_dim2=0
  g1[5] = 64u;                                 // tensor_dim0_stride = 64 (lo 32)
  g1[6] = 0u;                                  // stride hi | dim1_stride lo
  g1[7] = 0u;
  asm volatile("tensor_load_to_lds %0, %1" :: "s"(g0), "s"(g1) : "memory");
}

// ---------------------------------------------------------------------------
// Kernel 1: row squared norms
// ---------------------------------------------------------------------------
__global__ void sqnorm_kernel(const float* __restrict__ x,
                              float* __restrict__ sq, int n) {
  int i = blockIdx.x * blockDim.x + threadIdx.x;
  if (i >= n) return;
  const float4* p = (const float4*)(x + (size_t)i * D_IN);
  float s = 0.f;
#pragma unroll
  for (int j = 0; j < D_IN / 4; ++j) {
    float4 v = p[j];
    s += v.x * v.x + v.y * v.y + v.z * v.z + v.w * v.w;
  }
  sq[i] = s;
}

// ---------------------------------------------------------------------------
// Kernel 2: xfeat[N][1280] = [x@W_l + b_l | x@W_r + b_r] via fp32 WMMA.
// Block: 128 thr (4 waves). Block tile: 16 rows x 64 cols (16x16 per wave).
// grid = (N/16, 1280/64)
// ---------------------------------------------------------------------------
__global__ void __launch_bounds__(128)
feat_kernel(const float* __restrict__ x,
            const float* __restrict__ W_l, const float* __restrict__ b_l,
            const float* __restrict__ W_r, const float* __restrict__ b_r,
            float* __restrict__ xfeat) {
  __shared__ float Qs[16][68];                 // stride 68: conflict-free b64 reads

  const int tid  = threadIdx.x;
  const int wave = tid >> 5;
  const int lane = tid & 31;
  const int half = lane >> 4;                  // 0: K pair {0,1}; 1: {2,3}
  const int l16  = lane & 15;
  const int rowBase = blockIdx.x * 16;
  const int by      = blockIdx.y;              // 0..19; <10 -> W_l, else W_r

  const float* W  = (by < 10) ? W_l : W_r;
  const float* bv = (by < 10) ? b_l : b_r;
  const int colW  = ((by < 10) ? by : by - 10) * 64 + wave * 16 + l16; // in 0..639

  // stage 16x64 x-tile in LDS (coalesced float4)
  for (int i = tid; i < 16 * (D_IN / 4); i += 128) {
    int r = i >> 4, c4 = i & 15;
    float4 v = ((const float4*)(x + (size_t)(rowBase + r) * D_IN))[c4];
    Qs[r][c4 * 4 + 0] = v.x; Qs[r][c4 * 4 + 1] = v.y;
    Qs[r][c4 * 4 + 2] = v.z; Qs[r][c4 * 4 + 3] = v.w;
  }
  __syncthreads();

  v8f acc = {};
#pragma unroll
  for (int ks = 0; ks < D_IN / 4; ++ks) {
    const int kb = ks * 4 + 2 * half;
    v2f a = *(const v2f*)&Qs[l16][kb];         // A: M=l16, K=kb..kb+1
    v2f b;                                     // B: N=l16 (col), K=kb..kb+1
    b.x = W[(size_t)kb * F_TOT + colW];
    b.y = W[(size_t)(kb + 1) * F_TOT + colW];
    acc = wmma_f32_k4(a, b, acc);
  }

  const float bb = bv[colW];
  const int outCol = by * 64 + wave * 16 + l16;
#pragma unroll
  for (int r = 0; r < 8; ++r) {
    int row = rowBase + r + 8 * half;
    xfeat[(size_t)row * F2_TOT + outCol] = acc[r] + bb;
  }
}

// ---------------------------------------------------------------------------
// Kernel 3: fused KNN. Block: 128 thr (4 waves) owns 16 query rows.
// Loop over 256 tiles of 64 candidate cols, TDM double-buffered:
//   - wave 0 issues the DMA for tile ct+1, then s_wait_tensorcnt(1): TDM ops
//     complete in order per wave, so <=1 outstanding ==> tile ct has landed
//     while tile ct+1 streams in behind the current iteration's compute.
//   - WMMA 16x16 scores per wave (K=64 in 16 steps)
//   - score = sq_j - 2*dot (rank-equivalent to d^2), self -> +inf
//   - each thread keeps a sorted top-16 of its (row, col/8) stripe in regs
// Final: 8 partial lists per row merged exactly via LDS selection.
// Buffer-overwrite safety: Cs[buf^1] was last read in iteration ct-1's compute
// phase; every wave completes those reads before the mid-iteration barrier, and
// wave 0 issues the new DMA only after passing it, so no reader can be live.
// ---------------------------------------------------------------------------
__global__ void __launch_bounds__(128)
knn_kernel(const float* __restrict__ x, const float* __restrict__ sq,
           int* __restrict__ knn_out) {
  __shared__ float Qs[16][68];
  __shared__ float Cs[2][64][68];              // TDM double buffer
  __shared__ float Ds[16][68];
  __shared__ float sqs[64];
  __shared__ float md[16][128];
  __shared__ int   mi[16][128];

  const int tid  = threadIdx.x;
  const int wave = tid >> 5;
  const int lane = tid & 31;
  const int half = lane >> 4;
  const int l16  = lane & 15;
  const int rowBase = blockIdx.x * 16;

  const int myrow = tid & 15;                  // selection role
  const int sub   = tid >> 4;                  // 0..7

  const unsigned int cs_lds0 = (unsigned int)(uintptr_t)&Cs[0][0][0];
  const unsigned int cs_lds1 = (unsigned int)(uintptr_t)&Cs[1][0][0];

  // prefetch tile 0 into buffer 0 (overlaps with Qs staging below)
  if (wave == 0) tdm_load_tile_64x64(x, cs_lds0);

  // stage the 16x64 query tile once
  for (int i = tid; i < 16 * (D_IN / 4); i += 128) {
    int r = i >> 4, c4 = i & 15;
    float4 v = ((const float4*)(x + (size_t)(rowBase + r) * D_IN))[c4];
    Qs[r][c4 * 4 + 0] = v.x; Qs[r][c4 * 4 + 1] = v.y;
    Qs[r][c4 * 4 + 2] = v.z; Qs[r][c4 * 4 + 3] = v.w;
  }

  float bd[KNN_K]; int bi[KNN_K];              // sorted ascending, regs only
#pragma unroll
  for (int j = 0; j < KNN_K; ++j) { bd[j] = INFINITY; bi[j] = -1; }

  const int NT = N_NODES / 64;
  for (int ct = 0; ct < NT; ++ct) {
    const int buf = ct & 1;
    if (wave == 0) {
      if (ct + 1 < NT) {
        // kick off next tile into the other buffer, then ensure current landed
        tdm_load_tile_64x64(x + (size_t)(ct + 1) * 64 * D_IN,
                            buf ? cs_lds0 : cs_lds1);
        __builtin_amdgcn_s_wait_tensorcnt(1);
      } else {
        __builtin_amdgcn_s_wait_tensorcnt(0);
      }
    }
    if (tid < 64) sqs[tid] = sq[ct * 64 + tid];
    __syncthreads();                           // B2: tile ct + sqs visible

    // wave's 16x16 score tile: cols = ct*64 + wave*16 + (0..15)
    v8f acc = {};
#pragma unroll
    for (int ks = 0; ks < D_IN / 4; ++ks) {
      const int kb = ks * 4 + 2 * half;
      v2f a = *(const v2f*)&Qs[l16][kb];
      v2f b = *(const v2f*)&Cs[buf][wave * 16 + l16][kb];
      acc = wmma_f32_k4(a, b, acc);
    }
    const int lcol = wave * 16 + l16;
    const int gcol = ct * 64 + lcol;
    const float sj = sqs[lcol];
#pragma unroll
    for (int r = 0; r < 8; ++r) {
      int row = r + 8 * half;
      float s = sj - 2.0f * acc[r];
      if (gcol == rowBase + row) s = INFINITY;  // exclude self-loop
      Ds[row][lcol] = s;
    }
    __syncthreads();                           // B1: Ds ready

    // streaming top-16 update; common path is 1 compare (threshold miss)
#pragma unroll
    for (int c = 0; c < 8; ++c) {
      const int col = sub * 8 + c;
      float s = Ds[myrow][col];
      if (s < bd[KNN_K - 1]) {
        float cd = s; int ci = ct * 64 + col;
#pragma unroll
        for (int j = 0; j < KNN_K; ++j) {      // static-index insertion ladder
          if (cd < bd[j]) {
            float td = bd[j]; int ti = bi[j];
            bd[j] = cd; bi[j] = ci; cd = td; ci = ti;
          }
        }
      }
    }
  }

  // exact merge: union of 8 partial top-16 lists contains the true top-16
#pragma unroll
  for (int j = 0; j < KNN_K; ++j) {
    md[myrow][sub * KNN_K + j] = bd[j];
    mi[myrow][sub * KNN_K + j] = bi[j];
  }
  __syncthreads();
  if (tid < 16) {
    for (int j = 0; j < KNN_K; ++j) {
      float best = INFINITY; int barg = 0;
      for (int c = 0; c < 128; ++c) {
        float v = md[tid][c];
        if (v < best) { best = v; barg = c; }
      }
      knn_out[(size_t)(rowBase + tid) * KNN_K + j] = mi[tid][barg];
      md[tid][barg] = INFINITY;
    }
  }
}

// ---------------------------------------------------------------------------
// Kernel 4: GATv2 attention + aggregate. One wave32 per node (8 waves/block).
// Per head: cache x_r/att chunk in regs, logits via leaky_relu dot +
// butterfly reduce (all lanes get the sum -> uniform softmax over K=16),
// then alpha-weighted gather-sum. Gathers hit L2 (xfeat is 84MB).
// ---------------------------------------------------------------------------
__global__ void __launch_bounds__(256)
gat_kernel(const float* __restrict__ xfeat, const int* __restrict__ knn_in,
           const float* __restrict__ att, const float* __restrict__ bias,
           float* __restrict__ out) {
  const int lane = threadIdx.x & 31;
  const int node = blockIdx.x * 8 + (threadIdx.x >> 5);
  const float* xl = xfeat;                     // cols [0,640)
  const float* xr = xfeat + F_TOT;             // cols [640,1280)

  int nbr[KNN_K];
#pragma unroll
  for (int k = 0; k < KNN_K; ++k) nbr[k] = knn_in[(size_t)node * KNN_K + k];

  float hs0 = 0.f, hs1 = 0.f, hs2 = 0.f, hs3 = 0.f;

  for (int h = 0; h < HEADS; ++h) {
    float rv[4], av[4];
#pragma unroll
    for (int j = 0; j < 4; ++j) {
      rv[j] = xr[(size_t)node * F2_TOT + h * OUT_C + lane + 32 * j];
      av[j] = att[h * OUT_C + lane + 32 * j];
    }
    float lg[KNN_K];
    float mx = -INFINITY;
#pragma unroll
    for (int k = 0; k < KNN_K; ++k) {
      const float* mrow = xl + (size_t)nbr[k] * F2_TOT + h * OUT_C;
      float s = 0.f;
#pragma unroll
      for (int j = 0; j < 4; ++j) {
        float e = mrow[lane + 32 * j] + rv[j];
        e = (e > 0.f) ? e : NEG_SLOPE * e;     // leaky_relu
        s += e * av[j];
      }
#pragma unroll
      for (int m = 1; m < 32; m <<= 1) s += __shfl_xor(s, m, 32);
      lg[k] = s;                               // uniform across lanes
      mx = fmaxf(mx, s);
    }
    float den = 0.f;
#pragma unroll
    for (int k = 0; k < KNN_K; ++k) { lg[k] = __expf(lg[k] - mx); den += lg[k]; }
    const float inv = 1.0f / den;

    float a0 = 0.f, a1 = 0.f, a2 = 0.f, a3 = 0.f;
#pragma unroll
    for (int k = 0; k < KNN_K; ++k) {
      const float al = lg[k] * inv;
      const float* mrow = xl + (size_t)nbr[k] * F2_TOT + h * OUT_C;
      a0 += al * mrow[lane];
      a1 += al * mrow[lane + 32];
      a2 += al * mrow[lane + 64];
      a3 += al * mrow[lane + 96];
    }
    hs0 += a0; hs1 += a1; hs2 += a2; hs3 += a3;
  }
  const float inv5 = 1.0f / (float)HEADS;
  out[(size_t)node * OUT_C + lane +  0] = hs0 * inv5 + bias[lane];
  out[(size_t)node * OUT_C + lane + 32] = hs1 * inv5 + bias[lane + 32];
  out[(size_t)node * OUT_C + lane + 64] = hs2 * inv5 + bias[lane + 64];
  out[(size_t)node * OUT_C + lane + 96] = hs3 * inv5 + bias[lane + 96];
}

// ---------------------------------------------------------------------------
extern "C" void kernel_launch(void* const* d_in, const int* in_sizes, int n_in,
                              void* d_out, int out_size, void* d_ws, size_t ws_size,
                              hipStream_t stream) {
  const float* x    = (const float*)d_in[0];
  const float* W_l  = (const float*)d_in[1];
  const float* b_l  = (const float*)d_in[2];
  const float* W_r  = (const float*)d_in[3];
  const float* b_r  = (const float*)d_in[4];
  const float* att  = (const float*)d_in[5];
  const float* bias = (const float*)d_in[6];
  float* out = (float*)d_out;

  const int n = in_sizes[0] / D_IN;            // 16384

  // workspace layout
  char* ws = (char*)d_ws;
  const size_t xfeat_bytes = (size_t)n * F2_TOT * sizeof(float);   // ~84 MB
  const size_t sq_bytes    = (size_t)n * sizeof(float);
  float* xfeat = (float*)ws;
  float* sq    = (float*)(ws + xfeat_bytes);
  int*   knn   = (int*)(ws + xfeat_bytes + sq_bytes);

  sqnorm_kernel<<<(n + 255) / 256, 256, 0, stream>>>(x, sq, n);
  {
    dim3 g(n / 16, F2_TOT / 64);
    feat_kernel<<<g, 128, 0, stream>>>(x, W_l, b_l, W_r, b_r, xfeat);
  }
  knn_kernel<<<n / 16, 128, 0, stream>>>(x, sq, knn);
  gat_kernel<<<n / 8, 256, 0, stream>>>(xfeat, knn, att, bias, out);
}